// PointGridLoss_8340826489574
// MI455X (gfx1250) — compile-verified
//
#include <hip/hip_runtime.h>
#include <hip/hip_bf16.h>

typedef __attribute__((ext_vector_type(2))) float v2f;
typedef __attribute__((ext_vector_type(4))) float v4f;
typedef __attribute__((ext_vector_type(8))) float v8f;

#define BATCH   32
#define HDIM    768
#define WDIM    768
#define CELLS   8
#define STEP    96          // 768 / 8
#define NPTS    4096
#define NCELLS  (CELLS * CELLS)          // 64
#define NBLK    (BATCH * NCELLS)         // 2048
#define F4_PER_TILE 2304                 // 96 rows * 24 float4
#define F4_PER_ROW  24

// ---------------------------------------------------------------------------
// Kernel 1: 96x96 block sums of the density map, accumulated on the matrix
// pipe. A = ones(16x4); grand-total(D) = 16 * sum(B) independent of B layout,
// so each lane feeds 2 streamed floats per WMMA into the B operand.
// ---------------------------------------------------------------------------
__global__ __launch_bounds__(256) void pool_sum_wmma(
    const float* __restrict__ dens, float* __restrict__ ws_pred) {
  const int blk  = blockIdx.x;         // [0, 2048)
  const int b    = blk >> 6;
  const int cell = blk & 63;
  const int cy   = cell >> 3;
  const int cx   = cell & 7;
  const float* base = dens + (size_t)b * HDIM * WDIM
                           + (size_t)cy * STEP * WDIM
                           + (size_t)cx * STEP;
  const int tid = threadIdx.x;

  v2f a = {1.0f, 1.0f};   // ones A-matrix (16x4 across the wave)
  v8f c = {};             // f32 16x16 accumulator (8 VGPRs)

#pragma unroll
  for (int k = 0; k < F4_PER_TILE / 256; ++k) {   // 9 uniform iterations
    const int i   = tid + k * 256;
    const int row = i / F4_PER_ROW;
    const int c4  = i % F4_PER_ROW;
    const v4f* p = (const v4f*)(base + (size_t)row * WDIM + c4 * 4);
    v4f v = __builtin_nontemporal_load(p);        // TH=NT streaming b128 load
    v2f b0 = {v.x, v.y};
    v2f b1 = {v.z, v.w};
    // 8 args: (neg_a, A, neg_b, B, c_mod, C, reuse_a, reuse_b)
    c = __builtin_amdgcn_wmma_f32_16x16x4_f32(false, a, false, b0,
                                              (short)0, c, false, false);
    c = __builtin_amdgcn_wmma_f32_16x16x4_f32(false, a, false, b1,
                                              (short)0, c, false, false);
  }

  // sum(D) over all 8 VGPRs x 32 lanes == 16 * (tile partial sum of this wave)
  float s = c[0] + c[1] + c[2] + c[3] + c[4] + c[5] + c[6] + c[7];
#pragma unroll
  for (int off = 16; off > 0; off >>= 1) s += __shfl_xor(s, off, 32);

  __shared__ float wsum[8];
  if ((tid & 31) == 0) wsum[tid >> 5] = s;
  __syncthreads();
  if (tid == 0) {
    float t = 0.0f;
#pragma unroll
    for (int w = 0; w < 8; ++w) t += wsum[w];
    ws_pred[blk] = t * (1.0f / 16.0f);
  }
}

// ---------------------------------------------------------------------------
// Kernel 2: ground-truth point histogram, one workgroup per batch image.
// LDS f32 bins (ds_add_f32); integer counts => exact & order-independent.
// ---------------------------------------------------------------------------
__global__ __launch_bounds__(256) void gt_hist(
    const float* __restrict__ pts, const int* __restrict__ img_h,
    const int* __restrict__ img_w, float* __restrict__ ws_gt) {
  __shared__ float hist[NCELLS];
  const int tid = threadIdx.x;
  const int b   = blockIdx.x;
  if (tid < NCELLS) hist[tid] = 0.0f;
  __syncthreads();

  const float sx = (float)WDIM / (float)img_w[0];
  const float sy = (float)HDIM / (float)img_h[0];

  for (int n = tid; n < NPTS; n += 256) {
    const float2 p = *(const float2*)(pts + ((size_t)b * NPTS + n) * 2);
    const float px = p.x * sx;
    const float py = p.y * sy;
    const bool valid = (px >= 0.0f) && (px < (float)(CELLS * STEP)) &&
                       (py >= 0.0f) && (py < (float)(CELLS * STEP));
    if (valid) {
      int ix = (int)floorf(px / (float)STEP);
      int iy = (int)floorf(py / (float)STEP);
      ix = min(max(ix, 0), CELLS - 1);
      iy = min(max(iy, 0), CELLS - 1);
      atomicAdd(&hist[iy * CELLS + ix], 1.0f);
    }
  }
  __syncthreads();
  if (tid < NCELLS) ws_gt[b * NCELLS + tid] = hist[tid];
}

// ---------------------------------------------------------------------------
// Kernel 3: final scalar  sum |pred - gt| / B  over 2048 cells.
// ---------------------------------------------------------------------------
__global__ __launch_bounds__(256) void final_loss(
    const float* __restrict__ ws_pred, const float* __restrict__ ws_gt,
    float* __restrict__ out) {
  const int tid = threadIdx.x;
  float s = 0.0f;
  for (int i = tid; i < NBLK; i += 256) s += fabsf(ws_pred[i] - ws_gt[i]);
#pragma unroll
  for (int off = 16; off > 0; off >>= 1) s += __shfl_xor(s, off, 32);
  __shared__ float wsum[8];
  if ((tid & 31) == 0) wsum[tid >> 5] = s;
  __syncthreads();
  if (tid == 0) {
    float t = 0.0f;
#pragma unroll
    for (int w = 0; w < 8; ++w) t += wsum[w];
    out[0] = t / (float)BATCH;
  }
}

// ---------------------------------------------------------------------------
extern "C" void kernel_launch(void* const* d_in, const int* in_sizes, int n_in,
                              void* d_out, int out_size, void* d_ws,
                              size_t ws_size, hipStream_t stream) {
  const float* dens  = (const float*)d_in[0];   // [32,1,768,768] f32
  const float* pts   = (const float*)d_in[1];   // [32,4096,2]    f32
  const int*   img_h = (const int*)d_in[2];     // scalar
  const int*   img_w = (const int*)d_in[3];     // scalar
  float* out = (float*)d_out;

  float* ws_pred = (float*)d_ws;          // 2048 floats
  float* ws_gt   = ws_pred + NBLK;        // 2048 floats

  pool_sum_wmma<<<NBLK, 256, 0, stream>>>(dens, ws_pred);
  gt_hist<<<BATCH, 256, 0, stream>>>(pts, img_h, img_w, ws_gt);
  final_loss<<<1, 256, 0, stream>>>(ws_pred, ws_gt, out);
}